// MModalModel_43087111913541
// MI455X (gfx1250) — compile-verified
//
#include <hip/hip_runtime.h>
#include <hip/hip_bf16.h>

typedef __attribute__((ext_vector_type(2))) float v2f;
typedef __attribute__((ext_vector_type(8))) float v8f;

#define GN   50000
#define GE   640000
#define GFIN 100
#define GD   128
#define GH   8
#define GDH  16
#define GL   4
#define GDFF 512
#define GG   64
#define LDSROW 516   // padded row stride (floats) for 16x512 LDS tile

// ---------------------------------------------------------------- utilities
__global__ void zero_k(float* __restrict__ p, int n) {
  int t = blockIdx.x * blockDim.x + threadIdx.x;
  if (t < n) p[t] = 0.f;
}

// ------------------------------------------------ generic fp32 WMMA GEMM
// out[M x C] = A[M x K] @ B[K x C] (+bias) (+resid). One wave = one 16x16
// tile; blockDim = (32, C/16); gridDim.x = M/16. M and C multiples of 16,
// K multiple of 4 (true for all call sites: K in {100,128}, C in {128,256}).
__global__ __launch_bounds__(512) void gemm_wmma_k(
    const float* __restrict__ A, const float* __restrict__ B,
    const float* __restrict__ bias, const float* __restrict__ resid,
    float* __restrict__ out, int K, int lda, int ldb, int ldc, int ldr) {
  const int lane = threadIdx.x;
  const int l15  = lane & 15;
  const int hi   = lane >> 4;                 // 0: K+{0,1}, 1: K+{2,3}
  const int rowBase = blockIdx.x * 16;
  const int colBase = threadIdx.y * 16;
  const float* Arow = A + (size_t)(rowBase + l15) * lda;
  v8f acc = {};
  for (int k0 = 0; k0 < K; k0 += 4) {
    const int ka = k0 + hi * 2;
    v2f a = *(const v2f*)(Arow + ka);                       // A[M=l15][ka..ka+1]
    v2f b;
    b.x = B[(size_t)ka       * ldb + colBase + l15];        // B[ka  ][N=l15]
    b.y = B[(size_t)(ka + 1) * ldb + colBase + l15];        // B[ka+1][N=l15]
    acc = __builtin_amdgcn_wmma_f32_16x16x4_f32(false, a, false, b,
                                                (short)0, acc, false, false);
  }
  const int col = colBase + l15;
  const float bb = bias ? bias[col] : 0.f;
#pragma unroll
  for (int r = 0; r < 8; ++r) {
    const int row = rowBase + r + hi * 8;
    float vv = acc[r] + bb;
    if (resid) vv += resid[(size_t)row * ldr + col];
    out[(size_t)row * ldc + col] = vv;
  }
}

// ------------------------------------------- fused FFN: h += relu(hW1+b1)W2+b2
// One block = 16 rows of h. Phase 1: 16x512 intermediate -> LDS (relu).
// Phase 2: 16x128 output, A streamed from LDS. 8 waves / 256 threads.
__global__ __launch_bounds__(256) void ffn_fused_k(
    float* __restrict__ h, const float* __restrict__ W1,
    const float* __restrict__ b1, const float* __restrict__ W2,
    const float* __restrict__ b2) {
  __shared__ float tmp[16 * LDSROW];
  const int lane = threadIdx.x & 31;
  const int wave = threadIdx.x >> 5;          // 0..7
  const int l15  = lane & 15;
  const int hi   = lane >> 4;
  const int rowBase = blockIdx.x * 16;
  const float* Arow = h + (size_t)(rowBase + l15) * GD;

  for (int ct = wave; ct < GDFF / 16; ct += 8) {   // 32 column tiles
    v8f acc = {};
    const int col = ct * 16 + l15;
    for (int k0 = 0; k0 < GD; k0 += 4) {
      const int ka = k0 + hi * 2;
      v2f a = *(const v2f*)(Arow + ka);
      v2f b;
      b.x = W1[(size_t)ka       * GDFF + col];
      b.y = W1[(size_t)(ka + 1) * GDFF + col];
      acc = __builtin_amdgcn_wmma_f32_16x16x4_f32(false, a, false, b,
                                                  (short)0, acc, false, false);
    }
    const float bb = b1[col];
#pragma unroll
    for (int r = 0; r < 8; ++r)
      tmp[(r + hi * 8) * LDSROW + col] = fmaxf(acc[r] + bb, 0.f);
  }
  __syncthreads();
  {
    v8f acc = {};
    const int col = wave * 16 + l15;
    const float* Trow = tmp + (size_t)l15 * LDSROW;
    for (int k0 = 0; k0 < GDFF; k0 += 4) {
      const int ka = k0 + hi * 2;
      v2f a = *(const v2f*)(Trow + ka);
      v2f b;
      b.x = W2[(size_t)ka       * GD + col];
      b.y = W2[(size_t)(ka + 1) * GD + col];
      acc = __builtin_amdgcn_wmma_f32_16x16x4_f32(false, a, false, b,
                                                  (short)0, acc, false, false);
    }
    const float bb = b2[col];
#pragma unroll
    for (int r = 0; r < 8; ++r) {
      const int row = rowBase + r + hi * 8;
      h[(size_t)row * GD + col] += acc[r] + bb;
    }
  }
}

// ----------------------------------------------------- edge attention passes
__device__ __forceinline__ unsigned enc_f32(float f) {
  unsigned u = __float_as_uint(f);
  return (u & 0x80000000u) ? ~u : (u | 0x80000000u);
}
__device__ __forceinline__ float dec_f32(unsigned k) {
  return (k & 0x80000000u) ? __uint_as_float(k & 0x7FFFFFFFu)
                           : __uint_as_float(~k);
}

__global__ void edge_logits_max_k(const long long* __restrict__ ei,
                                  const float* __restrict__ qk,
                                  float* __restrict__ logits,
                                  unsigned* __restrict__ mkey, int EH) {
  int t = blockIdx.x * blockDim.x + threadIdx.x;
  if (t >= EH) return;
  const int e = t >> 3, hh = t & 7;
  const long long s = ei[e], d = ei[GE + e];
  const float4* qv = (const float4*)(qk + (size_t)s * (2 * GD) + hh * GDH);
  const float4* kv = (const float4*)(qk + (size_t)d * (2 * GD) + GD + hh * GDH);
  float acc = 0.f;
#pragma unroll
  for (int i = 0; i < 4; ++i) {
    float4 a = qv[i], b = kv[i];
    acc += a.x * b.x + a.y * b.y + a.z * b.z + a.w * b.w;
  }
  acc *= 0.25f;  // DH^-0.5
  logits[t] = acc;
  atomicMax(&mkey[(size_t)d * GH + hh], enc_f32(acc));
}

__global__ void edge_exp_sum_k(const long long* __restrict__ ei,
                               float* __restrict__ logits,
                               const unsigned* __restrict__ mkey,
                               float* __restrict__ ssum, int EH) {
  int t = blockIdx.x * blockDim.x + threadIdx.x;
  if (t >= EH) return;
  const int e = t >> 3, hh = t & 7;
  const long long d = ei[GE + e];
  const float m = dec_f32(mkey[(size_t)d * GH + hh]);
  const float p = expf(logits[t] - m);
  logits[t] = p;
  atomicAdd(&ssum[(size_t)d * GH + hh], p);
}

__global__ void edge_scatter_k(const long long* __restrict__ ei,
                               const float* __restrict__ logits,
                               const float* __restrict__ ssum,
                               const float* __restrict__ v,
                               float* __restrict__ agg, int EH) {
  int t = blockIdx.x * blockDim.x + threadIdx.x;
  if (t >= EH) return;
  const int e = t >> 3, hh = t & 7;
  const long long s = ei[e], d = ei[GE + e];
  const float attn = logits[t] / ssum[(size_t)d * GH + hh];
  const float4* vv = (const float4*)(v + (size_t)s * GD + hh * GDH);
  float* ag = agg + (size_t)d * GD + hh * GDH;
#pragma unroll
  for (int i = 0; i < 4; ++i) {
    float4 val = vv[i];
    atomicAdd(ag + i * 4 + 0, val.x * attn);
    atomicAdd(ag + i * 4 + 1, val.y * attn);
    atomicAdd(ag + i * 4 + 2, val.z * attn);
    atomicAdd(ag + i * 4 + 3, val.w * attn);
  }
}

// ---------------------------------------------------------------- pooling
__global__ void pool_cnt_k(const long long* __restrict__ batch,
                           float* __restrict__ cnt, int n) {
  int t = blockIdx.x * blockDim.x + threadIdx.x;
  if (t < n) atomicAdd(&cnt[batch[t]], 1.0f);
}
__global__ void pool_add_k(const long long* __restrict__ batch,
                           const float* __restrict__ h,
                           float* __restrict__ pooled, int n) {
  int t = blockIdx.x * blockDim.x + threadIdx.x;
  if (t >= n) return;
  const int nidx = t >> 7, dcol = t & 127;
  atomicAdd(&pooled[(size_t)batch[nidx] * GD + dcol], h[t]);
}
__global__ void pool_fin_k(const float* __restrict__ pooled,
                           const float* __restrict__ cnt,
                           float* __restrict__ out, int n) {
  int t = blockIdx.x * blockDim.x + threadIdx.x;
  if (t < n) out[t] = pooled[t] / fmaxf(cnt[t >> 7], 1.0f);
}

// ---------------------------------------------------------------- launcher
extern "C" void kernel_launch(void* const* d_in, const int* in_sizes, int n_in,
                              void* d_out, int out_size, void* d_ws,
                              size_t ws_size, hipStream_t stream) {
  (void)in_sizes; (void)n_in; (void)out_size; (void)ws_size;
  const float*     x       = (const float*)d_in[0];
  const long long* ei      = (const long long*)d_in[1];
  const long long* batch   = (const long long*)d_in[2];
  const float*     W_embed = (const float*)d_in[3];
  const float*     Wqk     = (const float*)d_in[4];
  const float*     Wv      = (const float*)d_in[5];
  const float*     Wout    = (const float*)d_in[6];
  const float*     b_out   = (const float*)d_in[7];
  const float*     W1      = (const float*)d_in[8];
  const float*     b1      = (const float*)d_in[9];
  const float*     W2      = (const float*)d_in[10];
  const float*     b2      = (const float*)d_in[11];
  float* ws = (float*)d_ws;

  // workspace layout (floats)
  float*    h      = ws;                       // 6,400,000
  float*    qkb    = ws + 6400000;             // 12,800,000
  float*    vb     = ws + 19200000;            // 6,400,000
  float*    pbuf   = ws + 25600000;            // 5,120,000 (logits -> p)
  float*    agg    = ws + 30720000;            // 6,400,000
  unsigned* mkey   = (unsigned*)(ws + 37120000); // 400,000
  float*    ssum   = ws + 37520000;            // 400,000
  float*    pooled = ws + 37920000;            // 8,192
  float*    cnt    = ws + 37928192;            // 64

  const int rowTiles = GN / 16;                // 3125 (exact)
  const int EH = GE * GH;                      // 5,120,000
  const int ehBlocks = (EH + 255) / 256;

  // h = x @ W_embed   (K=100, C=128)
  gemm_wmma_k<<<dim3(rowTiles), dim3(32, 8), 0, stream>>>(
      x, W_embed, nullptr, nullptr, h, GFIN, GFIN, GD, GD, GD);

  for (int l = 0; l < GL; ++l) {
    const float* Wqk_l  = Wqk  + (size_t)l * GD * 2 * GD;
    const float* Wv_l   = Wv   + (size_t)l * GD * GD;
    const float* Wout_l = Wout + (size_t)l * GD * GD;
    const float* bo_l   = b_out + (size_t)l * GD;
    const float* W1_l   = W1 + (size_t)l * GD * GDFF;
    const float* b1_l   = b1 + (size_t)l * GDFF;
    const float* W2_l   = W2 + (size_t)l * GDFF * GD;
    const float* b2_l   = b2 + (size_t)l * GD;

    // qk = h @ Wqk[l]  (C=256),  v = h @ Wv[l]  (C=128)
    gemm_wmma_k<<<dim3(rowTiles), dim3(32, 16), 0, stream>>>(
        h, Wqk_l, nullptr, nullptr, qkb, GD, GD, 2 * GD, 2 * GD, 0);
    gemm_wmma_k<<<dim3(rowTiles), dim3(32, 8), 0, stream>>>(
        h, Wv_l, nullptr, nullptr, vb, GD, GD, GD, GD, 0);

    // zero agg + mkey + ssum (contiguous: 7,200,000 floats)
    zero_k<<<dim3((7200000 + 255) / 256), dim3(256), 0, stream>>>(agg, 7200000);

    edge_logits_max_k<<<dim3(ehBlocks), dim3(256), 0, stream>>>(
        ei, qkb, pbuf, mkey, EH);
    edge_exp_sum_k<<<dim3(ehBlocks), dim3(256), 0, stream>>>(
        ei, pbuf, mkey, ssum, EH);
    edge_scatter_k<<<dim3(ehBlocks), dim3(256), 0, stream>>>(
        ei, pbuf, ssum, vb, agg, EH);

    // h = h + agg @ Wout[l] + b_out[l]   (in-place safe: elementwise resid)
    gemm_wmma_k<<<dim3(rowTiles), dim3(32, 8), 0, stream>>>(
        agg, Wout_l, bo_l, h, h, GD, GD, GD, GD, GD);

    // h += relu(h @ W1 + b1) @ W2 + b2   (fused, LDS intermediate)
    ffn_fused_k<<<dim3(rowTiles), dim3(256), 0, stream>>>(
        h, W1_l, b1_l, W2_l, b2_l);
  }

  // pooled mean over batch groups
  zero_k<<<dim3((8256 + 255) / 256), dim3(256), 0, stream>>>(pooled, 8256);
  pool_cnt_k<<<dim3((GN + 255) / 256), dim3(256), 0, stream>>>(batch, cnt, GN);
  pool_add_k<<<dim3((GN * GD + 255) / 256), dim3(256), 0, stream>>>(
      batch, h, pooled, GN * GD);
  pool_fin_k<<<dim3((GG * GD + 255) / 256), dim3(256), 0, stream>>>(
      pooled, cnt, (float*)d_out, GG * GD);
}